// LocalChargeEquilibrationBlock_13872744366327
// MI455X (gfx1250) — compile-verified
//
#include <hip/hip_runtime.h>
#include <hip/hip_bf16.h>

#define BB 256
#define NN 64
#define EE 512
#define NELEM 10
#define MP 528   // augmented dim 513 padded to multiple of 16
#define LDA 528

typedef __attribute__((ext_vector_type(2))) float v2f;
typedef __attribute__((ext_vector_type(8))) float v8f;

__device__ __forceinline__ v8f wmma_f32_16x16x4(v2f a, v2f b, v8f c) {
  // 8-arg pattern: (neg_a, A, neg_b, B, c_mod, C, reuse_a, reuse_b)
  return __builtin_amdgcn_wmma_f32_16x16x4_f32(false, a, false, b, (short)0, c,
                                               false, false);
}

// ---------------------------------------------------------------------------
// Kernel 1: per-molecule 64x64 "A" matrix (erf kernel + hardness diagonal)
// ---------------------------------------------------------------------------
__global__ void build_A_kernel(const float* __restrict__ dists,
                               const float* __restrict__ attrs,
                               const int*   __restrict__ an,
                               const float* __restrict__ hardness,
                               const float* __restrict__ cov_radii,
                               float* __restrict__ Aout) {
  const int b = blockIdx.x;
  const int i = threadIdx.x;          // 0..63
  __shared__ float sig_s[NN];
  __shared__ float diag_s[NN];
  float c = cov_radii[an[b * NN + i]];
  const float* at = attrs + ((size_t)b * NN + i) * NELEM;
  int best = 0; float bv = at[0];
#pragma unroll
  for (int k = 1; k < NELEM; ++k) { float v = at[k]; if (v > bv) { bv = v; best = k; } }
  float h = hardness[best];
  const float inv_sqrt_pi = 0.5641895835477563f;   // 1/sqrt(pi)
  diag_s[i] = h + inv_sqrt_pi / c;
  sig_s[i]  = c * c;
  __syncthreads();
  const float inv_sqrt2 = 0.7071067811865476f;
  const float* drow = dists + ((size_t)b * NN + i) * NN;
  float*       arow = Aout  + ((size_t)b * NN + i) * NN;
  float si = sig_s[i];
  for (int j = 0; j < NN; ++j) {
    float d = drow[j];
    float g = sqrtf(si + sig_s[j]);
    arow[j] = (i == j) ? diag_s[i] : erff(d * inv_sqrt2 / g) / d;
  }
}

// ---------------------------------------------------------------------------
// Kernel 2: W = A @ T   (64x64 * 64x512), WMMA f32 16x16x4, 1 wave per tile
// ---------------------------------------------------------------------------
__global__ void gemm_W_kernel(const float* __restrict__ Amat,
                              const float* __restrict__ T,
                              float* __restrict__ W) {
  const int b    = blockIdx.x;
  const int wave = threadIdx.x >> 5;
  const int lane = threadIdx.x & 31;
  const int half = lane >> 4;
  const int mr   = lane & 15;
  const int t  = blockIdx.y * 4 + wave;  // 0..127 : 4 row-tiles x 32 col-tiles
  const int m0 = (t >> 5) * 16;
  const int n0 = (t & 31) * 16;
  const float* Ab = Amat + (size_t)b * NN * NN;
  const float* Tb = T    + (size_t)b * NN * EE;
  v8f acc = {0.f,0.f,0.f,0.f,0.f,0.f,0.f,0.f};
  for (int k0 = 0; k0 < NN; k0 += 4) {
    v2f a;
    const float* ap = Ab + (size_t)(m0 + mr) * NN + k0 + 2 * half;
    a.x = ap[0]; a.y = ap[1];
    v2f bf;
    bf.x = Tb[(size_t)(k0 + 2 * half + 0) * EE + n0 + mr];
    bf.y = Tb[(size_t)(k0 + 2 * half + 1) * EE + n0 + mr];
    acc = wmma_f32_16x16x4(a, bf, acc);
  }
  float* Wb = W + (size_t)b * NN * EE;
#pragma unroll
  for (int r = 0; r < 8; ++r)
    Wb[(size_t)(m0 + r + 8 * half) * EE + n0 + mr] = acc[r];
}

// ---------------------------------------------------------------------------
// Kernel 3: Aee = T^T @ W  (512x64 * 64x512) into padded Aaug (lda 528)
// ---------------------------------------------------------------------------
__global__ void gemm_Aee_kernel(const float* __restrict__ T,
                                const float* __restrict__ W,
                                float* __restrict__ Aaug) {
  const int b    = blockIdx.x;
  const int wave = threadIdx.x >> 5;
  const int lane = threadIdx.x & 31;
  const int half = lane >> 4;
  const int mr   = lane & 15;
  const int t  = blockIdx.y * 4 + wave;  // 0..1023 : 32x32 tiles
  const int m0 = (t >> 5) * 16;
  const int n0 = (t & 31) * 16;
  const float* Tb = T + (size_t)b * NN * EE;
  const float* Wb = W + (size_t)b * NN * EE;
  float* Cb = Aaug + (size_t)b * MP * LDA;
  v8f acc = {0.f,0.f,0.f,0.f,0.f,0.f,0.f,0.f};
  for (int k0 = 0; k0 < NN; k0 += 4) {
    v2f a;   // A[m][k] = T[k][m0+m]
    a.x = Tb[(size_t)(k0 + 2 * half + 0) * EE + m0 + mr];
    a.y = Tb[(size_t)(k0 + 2 * half + 1) * EE + m0 + mr];
    v2f bf;
    bf.x = Wb[(size_t)(k0 + 2 * half + 0) * EE + n0 + mr];
    bf.y = Wb[(size_t)(k0 + 2 * half + 1) * EE + n0 + mr];
    acc = wmma_f32_16x16x4(a, bf, acc);
  }
#pragma unroll
  for (int r = 0; r < 8; ++r)
    Cb[(size_t)(m0 + r + 8 * half) * LDA + n0 + mr] = acc[r];
}

// ---------------------------------------------------------------------------
// Kernel 4: diagonal cutoffs, ones augmentation, identity padding, rhs b
// ---------------------------------------------------------------------------
__global__ void assemble_kernel(const float* __restrict__ T,
                                const float* __restrict__ eneg,
                                const float* __restrict__ dists,
                                const float* __restrict__ tq,
                                const int* __restrict__ edge_index,
                                float* __restrict__ Aaug,
                                float* __restrict__ bvec) {
  const int b   = blockIdx.x;
  const int tid = threadIdx.x;          // 0..255
  float* Ab = Aaug + (size_t)b * MP * LDA;
  float* bb = bvec + (size_t)b * MP;
  const float* Tb = T    + (size_t)b * NN * EE;
  const float* en = eneg + (size_t)b * NN;
  const float PIv = 3.14159265358979323846f;
  for (int e = tid; e < EE; e += 256) {
    int s = edge_index[(size_t)b * 2 * EE + e];
    int d = edge_index[(size_t)b * 2 * EE + EE + e];
    float el  = dists[(size_t)b * NN * NN + s * NN + d];
    float cut = 1.0f / cosf(PIv * el * 0.1f) - 1.0f;  // pi*el/(2*R_CUT), R_CUT=5
    Ab[(size_t)e * LDA + e] += cut;
  }
  // ones row/column of the augmented system
  for (int j = tid; j <= EE; j += 256) Ab[(size_t)EE * LDA + j] = 1.0f;
  for (int i = tid; i < EE; i += 256)  Ab[(size_t)i * LDA + EE] = 1.0f;
  // padding region -> identity block (pad unknowns solve to 0)
  const int PW = MP - EE - 1;  // 15
  for (int idx = tid; idx < MP * PW; idx += 256) {
    int i = idx / PW;
    int j = EE + 1 + idx % PW;
    Ab[(size_t)i * LDA + j] = (i == j) ? 1.0f : 0.0f;
  }
  for (int idx = tid; idx < PW * (EE + 1); idx += 256) {
    int i = EE + 1 + idx / (EE + 1);
    int j = idx % (EE + 1);
    Ab[(size_t)i * LDA + j] = 0.0f;
  }
  // rhs
  for (int e = tid; e < EE; e += 256) {
    float s = 0.f;
    for (int n = 0; n < NN; ++n) s += Tb[(size_t)n * EE + e] * en[n];
    bb[e] = -s;
  }
  if (tid == 0) bb[EE] = tq[b];
  for (int j = EE + 1 + tid; j < MP; j += 256) bb[j] = 0.f;
}

// ---------------------------------------------------------------------------
// Kernel 5: blocked LU (bs=16, no pivoting) with WMMA trailing update,
//           then forward/backward substitution. One 256-thread WG / molecule.
// ---------------------------------------------------------------------------
__global__ void lu_solve_kernel(float* __restrict__ Aaug, float* __restrict__ bvec) {
  const int b    = blockIdx.x;
  const int tid  = threadIdx.x;      // 0..255 (8 waves)
  const int wave = tid >> 5;
  const int lane = tid & 31;
  const int half = lane >> 4;
  const int mr   = lane & 15;
  float* Ab = Aaug + (size_t)b * MP * LDA;
  __shared__ float D[16][17];
  __shared__ float xs[MP];
  const int NBLK = MP / 16;          // 33
  const int di = tid >> 4, dj = tid & 15;

  for (int kb = 0; kb < NBLK; ++kb) {
    const int k0 = kb * 16;
    // ---- load + factor 16x16 diagonal block in LDS ----
    D[di][dj] = Ab[(size_t)(k0 + di) * LDA + k0 + dj];
    __syncthreads();
    for (int k = 0; k < 15; ++k) {
      if (dj == k && di > k) D[di][k] /= D[k][k];
      __syncthreads();
      if (di > k && dj > k) D[di][dj] -= D[di][k] * D[k][dj];
      __syncthreads();
    }
    Ab[(size_t)(k0 + di) * LDA + k0 + dj] = D[di][dj];
    // ---- U row panel: solve L11 * U = A(k,:) column-by-thread ----
    for (int j = k0 + 16 + tid; j < MP; j += 256) {
      float u[16];
#pragma unroll
      for (int i = 0; i < 16; ++i) u[i] = Ab[(size_t)(k0 + i) * LDA + j];
#pragma unroll
      for (int i = 1; i < 16; ++i) {
        float s = u[i];
        for (int t2 = 0; t2 < i; ++t2) s -= D[i][t2] * u[t2];
        u[i] = s;
      }
#pragma unroll
      for (int i = 0; i < 16; ++i) Ab[(size_t)(k0 + i) * LDA + j] = u[i];
    }
    // ---- L col panel: solve L * U11 = A(:,k) row-by-thread ----
    for (int i = k0 + 16 + tid; i < MP; i += 256) {
      float l[16];
#pragma unroll
      for (int j = 0; j < 16; ++j) l[j] = Ab[(size_t)i * LDA + k0 + j];
#pragma unroll
      for (int j = 0; j < 16; ++j) {
        float s = l[j];
        for (int t2 = 0; t2 < j; ++t2) s -= l[t2] * D[t2][j];
        l[j] = s / D[j][j];
      }
#pragma unroll
      for (int j = 0; j < 16; ++j) Ab[(size_t)i * LDA + k0 + j] = l[j];
    }
    __threadfence_block();
    __syncthreads();
    // ---- trailing update C -= L*U : 16x16 tiles, K=16 = 4 x wmma f32 16x16x4
    const int r = NBLK - 1 - kb;
    const int numTiles = r * r;
    for (int t = wave; t < numTiles; t += 8) {
      const int i0 = (kb + 1 + t / r) * 16;
      const int j0 = (kb + 1 + t % r) * 16;
      v8f acc;
#pragma unroll
      for (int rr = 0; rr < 8; ++rr)
        acc[rr] = Ab[(size_t)(i0 + rr + 8 * half) * LDA + j0 + mr];
#pragma unroll
      for (int kk = 0; kk < 16; kk += 4) {
        v2f a;  // negated L (fp32 WMMA has no A-negate, only C-negate)
        const float* lp = Ab + (size_t)(i0 + mr) * LDA + k0 + kk + 2 * half;
        a.x = -lp[0]; a.y = -lp[1];
        v2f bf;
        bf.x = Ab[(size_t)(k0 + kk + 2 * half + 0) * LDA + j0 + mr];
        bf.y = Ab[(size_t)(k0 + kk + 2 * half + 1) * LDA + j0 + mr];
        acc = wmma_f32_16x16x4(a, bf, acc);
      }
#pragma unroll
      for (int rr = 0; rr < 8; ++rr)
        Ab[(size_t)(i0 + rr + 8 * half) * LDA + j0 + mr] = acc[rr];
    }
    __threadfence_block();
    __syncthreads();
  }

  // ---- forward substitution (L unit-lower) ----
  for (int i = tid; i < MP; i += 256) xs[i] = bvec[(size_t)b * MP + i];
  __syncthreads();
  for (int k = 0; k < MP - 1; ++k) {
    float xk = xs[k];
    for (int i = k + 1 + tid; i < MP; i += 256)
      xs[i] -= Ab[(size_t)i * LDA + k] * xk;
    __syncthreads();
  }
  // ---- backward substitution (U) ----
  for (int k = MP - 1; k >= 0; --k) {
    if (tid == 0) xs[k] = xs[k] / Ab[(size_t)k * LDA + k];
    __syncthreads();
    float xk = xs[k];
    for (int i = tid; i < k; i += 256)
      xs[i] -= Ab[(size_t)i * LDA + k] * xk;
    __syncthreads();
  }
  for (int i = tid; i < MP; i += 256) bvec[(size_t)b * MP + i] = xs[i];
}

// ---------------------------------------------------------------------------
// Kernel 6: charges = T @ x[:E]
// ---------------------------------------------------------------------------
__global__ void charges_kernel(const float* __restrict__ T,
                               const float* __restrict__ xv,
                               float* __restrict__ out) {
  const int b = blockIdx.x;
  const int n = threadIdx.x;  // 0..63
  const float* Tb = T + (size_t)b * NN * EE + (size_t)n * EE;
  const float* xb = xv + (size_t)b * MP;
  float s = 0.f;
  for (int e = 0; e < EE; ++e) s += Tb[e] * xb[e];
  out[(size_t)b * NN + n] = s;
}

// ---------------------------------------------------------------------------
extern "C" void kernel_launch(void* const* d_in, const int* in_sizes, int n_in,
                              void* d_out, int out_size, void* d_ws, size_t ws_size,
                              hipStream_t stream) {
  (void)in_sizes; (void)n_in; (void)out_size; (void)ws_size;
  const float* T     = (const float*)d_in[0];   // [B,N,E]
  const float* eneg  = (const float*)d_in[1];   // [B,N]
  const float* dists = (const float*)d_in[2];   // [B,N,N]
  const float* attrs = (const float*)d_in[3];   // [B,N,10]
  const float* tq    = (const float*)d_in[4];   // [B]
  const float* hard  = (const float*)d_in[5];   // [10]
  const float* cov   = (const float*)d_in[6];   // [10]
  const int*   an    = (const int*)d_in[7];     // [B,N]
  const int*   ei    = (const int*)d_in[8];     // [B,2,E]

  float* ws   = (float*)d_ws;
  float* Amat = ws;                              // B*64*64
  float* W    = Amat + (size_t)BB * NN * NN;     // B*64*512
  float* Aaug = W    + (size_t)BB * NN * EE;     // B*528*528
  float* bvec = Aaug + (size_t)BB * MP * LDA;    // B*528

  build_A_kernel  <<<BB, NN,            0, stream>>>(dists, attrs, an, hard, cov, Amat);
  gemm_W_kernel   <<<dim3(BB, 32), 128, 0, stream>>>(Amat, T, W);
  gemm_Aee_kernel <<<dim3(BB, 256), 128, 0, stream>>>(T, W, Aaug);
  assemble_kernel <<<BB, 256,           0, stream>>>(T, eneg, dists, tq, ei, Aaug, bvec);
  lu_solve_kernel <<<BB, 256,           0, stream>>>(Aaug, bvec);
  charges_kernel  <<<BB, NN,            0, stream>>>(T, bvec, (float*)d_out);
}